// GCNConv_27616639713353
// MI455X (gfx1250) — compile-verified
//
#include <hip/hip_runtime.h>

typedef __attribute__((ext_vector_type(2))) float v2f;
typedef __attribute__((ext_vector_type(8))) float v8f;

#define D 128            // D_IN == D_OUT == 128

// ---------------------------------------------------------------------------
// Kernel 1: zero the aggregation workspace (float4 stores, coalesced)
// ---------------------------------------------------------------------------
__global__ void gcn_zero_kernel(float4* __restrict__ p, int n4) {
    int i = blockIdx.x * blockDim.x + threadIdx.x;
    if (i < n4) p[i] = make_float4(0.f, 0.f, 0.f, 0.f);
}

// ---------------------------------------------------------------------------
// Kernel 2: edge scatter-add.  One wave32 per edge: lane l handles dims
// [4l, 4l+4) -> one global_load_b128 + 4 hardware f32 atomics into L2.
// feature (25.6 MB) and agg (25.6 MB) both fit in the 192 MB L2.
// ---------------------------------------------------------------------------
__global__ void __launch_bounds__(256)
gcn_scatter_kernel(const float* __restrict__ feat,
                   const int* __restrict__ src,
                   const int* __restrict__ dst,
                   float* __restrict__ agg,
                   int nedges) {
    const int lane   = threadIdx.x & 31;
    const int wave   = (blockIdx.x * blockDim.x + threadIdx.x) >> 5;
    const int nwaves = (gridDim.x * blockDim.x) >> 5;

    for (int e = wave; e < nedges; e += nwaves) {
        const int s = src[e];          // wave-uniform -> scalarized by compiler
        const int d = dst[e];
        const float4 v = *(const float4*)(feat + (size_t)s * D + lane * 4);
        float* out = agg + (size_t)d * D + lane * 4;
        unsafeAtomicAdd(out + 0, v.x); // global_atomic_add_f32 (no CAS loop)
        unsafeAtomicAdd(out + 1, v.y);
        unsafeAtomicAdd(out + 2, v.z);
        unsafeAtomicAdd(out + 3, v.w);
    }
}

// ---------------------------------------------------------------------------
// Kernel 3: z = relu(agg @ W^T + b) + feature, via V_WMMA_F32_16X16X4_F32.
// Block = 256 threads = 8 waves. Block owns a 16-row tile; wave w owns the
// 16-column tile [16w, 16w+16) of the 128 outputs. K-loop: 32 WMMAs.
// ---------------------------------------------------------------------------
__global__ void __launch_bounds__(256)
gcn_gemm_kernel(const float* __restrict__ agg,
                const float* __restrict__ W,     // [D_OUT, D_IN] row-major
                const float* __restrict__ bias,  // [D_OUT]
                const float* __restrict__ feat,  // residual
                float* __restrict__ out,
                int nrows) {
    const int lane   = threadIdx.x & 31;
    const int wave   = threadIdx.x >> 5;          // 0..7 -> column tile
    const int tile_m = blockIdx.x * 16;
    const int col0   = wave * 16;

    const int half = lane >> 4;                   // 0: lanes 0-15, 1: lanes 16-31
    const int l15  = lane & 15;

    // A-fragment row for this lane (clamped for a possible ragged tail tile;
    // stores are guarded below). M = lane % 16 for both half-waves.
    const int row_a = min(tile_m + l15, nrows - 1);
    const float* arow = agg + (size_t)row_a * D + 2 * half;   // K offset 2*half

    // B-fragment: lane -> output column o = col0 + l15; K offset 2*half.
    const float* brow = W + (size_t)(col0 + l15) * D + 2 * half;

    v8f c = {};
#pragma unroll
    for (int k0 = 0; k0 < D; k0 += 4) {
        const v2f a = *(const v2f*)(arow + k0);   // A[K=k0+2h, k0+2h+1]
        const v2f b = *(const v2f*)(brow + k0);   // B[K=k0+2h, k0+2h+1]
        c = __builtin_amdgcn_wmma_f32_16x16x4_f32(
                /*neg_a=*/false, a, /*neg_b=*/false, b,
                /*c_mod=*/(short)0, c, /*reuse_a=*/false, /*reuse_b=*/false);
    }

    // Epilogue: bias + relu + residual.  C layout: VGPR r -> M = r + 8*half,
    // N = lane & 15 (lanes 0-15 and 16-31 both cover N=0..15).
    const int n  = col0 + l15;
    const float bn = bias[n];
#pragma unroll
    for (int r = 0; r < 8; ++r) {
        const int m   = r + 8 * half;
        const int row = tile_m + m;
        if (row < nrows) {
            float z = c[r] + bn;
            z = z > 0.f ? z : 0.f;
            out[(size_t)row * D + n] = z + feat[(size_t)row * D + n];
        }
    }
}

// ---------------------------------------------------------------------------
extern "C" void kernel_launch(void* const* d_in, const int* in_sizes, int n_in,
                              void* d_out, int out_size, void* d_ws, size_t ws_size,
                              hipStream_t stream) {
    const float* feature = (const float*)d_in[0];   // [N, 128] f32
    const int*   src     = (const int*)  d_in[1];   // [E] i32
    const int*   dst     = (const int*)  d_in[2];   // [E] i32
    const float* W       = (const float*)d_in[3];   // [128, 128] f32
    const float* b       = (const float*)d_in[4];   // [128] f32
    float*       out     = (float*)d_out;

    const int nnodes = in_sizes[0] / D;
    const int nedges = in_sizes[1];

    float* agg = (float*)d_ws;                      // [N, 128] f32 scratch

    // 1) agg = 0
    const int n4 = nnodes * (D / 4);
    gcn_zero_kernel<<<(n4 + 255) / 256, 256, 0, stream>>>((float4*)agg, n4);

    // 2) scatter-add over edges (16384 waves grid-striding 800k edges)
    gcn_scatter_kernel<<<2048, 256, 0, stream>>>(feature, src, dst, agg, nedges);

    // 3) fused WMMA GEMM + bias + relu + residual
    const int mtiles = (nnodes + 15) / 16;
    gcn_gemm_kernel<<<mtiles, 256, 0, stream>>>(agg, W, b, feature, out, nnodes);
}